// MultiScaleTrendLoss_910533067632
// MI455X (gfx1250) — compile-verified
//
#include <hip/hip_runtime.h>

typedef __attribute__((ext_vector_type(2))) float v2f;
typedef __attribute__((ext_vector_type(8))) float v8f;

#define BATCH 64
#define TLEN  4096
#define CH    64
#define NCHUNK 16
#define CHUNK (TLEN / NCHUNK)          // 256 timesteps accumulated per chunk
#define WARM  256                      // warm-up steps; 0.9^256 ~ 2e-12 (exact to fp32)
#define CTILES (CH / 16)               // 4 channel tiles of 16
#define NWAVES (BATCH * CTILES * NCHUNK)   // 4096 independent waves
#define BLOCK_THREADS 128
#define WAVES_PER_BLOCK (BLOCK_THREADS / 32)
#define NBLOCKS (NWAVES / WAVES_PER_BLOCK) // 1024 blocks

// One wave handles one (batch, 16-channel tile, time-chunk).
// Per 16x16 (t x c) tile of d = pred - target, compute for each alpha:
//   E = L_alpha @ D_tile + outer((1-a)^{m+1}, carry)
// via a chain of 4 V_WMMA_F32_16X16X4_F32, then accumulate w * E^2.
__global__ __launch_bounds__(BLOCK_THREADS)
void MultiScaleTrendLoss_ema_wmma_kernel(const float* __restrict__ pred,
                                         const float* __restrict__ target,
                                         float* __restrict__ partials)
{
    const int lane = threadIdx.x & 31;
    const int wib  = threadIdx.x >> 5;
    const int wid  = blockIdx.x * WAVES_PER_BLOCK + wib;   // [0, NWAVES)

    const int b     = wid >> 6;        // 64 batches
    const int rem   = wid & 63;
    const int ctile = rem & 3;         // 4 channel tiles
    const int chunk = rem >> 2;        // 16 time chunks

    const int n    = lane & 15;        // column (channel within tile) / A-row
    const int half = lane >> 4;        // lane half selects K/M offsets
    const int m    = n;                // A-matrix row for this lane
    const int c    = ctile * 16 + n;

    const float alphas[3]  = {0.1f, 0.3f, 0.5f};
    const float weights[3] = {0.5f, 0.3f, 0.2f};

    // A-matrix fragments (constant lower-triangular EMA operator) and carry decay.
    // A 16x4 layout: lane holds row m = lane%16; VGPR0 = col (4k + 2*half),
    // VGPR1 = col (4k + 2*half + 1).
    v2f Afr[3][4];
    v8f dec[3];
    for (int al = 0; al < 3; ++al) {
        const float a = alphas[al];
        const float r = 1.0f - a;
        float rp[17];
        rp[0] = 1.0f;
        #pragma unroll
        for (int j = 1; j <= 16; ++j) rp[j] = rp[j - 1] * r;
        #pragma unroll
        for (int k = 0; k < 4; ++k) {
            const int s0 = 4 * k + 2 * half;
            const int s1 = s0 + 1;
            Afr[al][k][0] = (s0 <= m) ? a * rp[m - s0] : 0.0f;
            Afr[al][k][1] = (s1 <= m) ? a * rp[m - s1] : 0.0f;
        }
        // C/D layout: VGPR i holds row (i + 8*half); carry decay = r^{row+1}
        #pragma unroll
        for (int i = 0; i < 8; ++i) dec[al][i] = rp[i + 8 * half + 1];
    }

    const size_t base = (size_t)b * TLEN * CH + (size_t)c;
    const float* pb = pred + base;
    const float* tb = target + base;

    const int t_acc   = chunk * CHUNK;
    const int t_start = (chunk == 0) ? 0 : (t_acc - WARM);
    const int t_end   = t_acc + CHUNK;

    // carry = d[t_start] makes the t=0 boundary exact and warms up interior chunks
    const float d0 = pb[(size_t)t_start * CH] - tb[(size_t)t_start * CH];
    float carry[3] = {d0, d0, d0};

    float acc = 0.0f;

    for (int t0 = t_start; t0 < t_end; t0 += 16) {
        // stream ahead (global_prefetch_b8); whole dataset is L2-resident (128MB < 192MB)
        if (t0 + 64 < t_end) {
            __builtin_prefetch(pb + (size_t)(t0 + 64) * CH, 0, 1);
            __builtin_prefetch(tb + (size_t)(t0 + 64) * CH, 0, 1);
        }

        // B 4x16 fragments: lane holds col n; v0 = row (t0+4k+2*half), v1 = next row
        v2f Bfr[4];
        #pragma unroll
        for (int k = 0; k < 4; ++k) {
            const size_t row = (size_t)(t0 + 4 * k + 2 * half) * CH;
            const float p0 = pb[row];
            const float p1 = pb[row + CH];
            const float q0 = tb[row];
            const float q1 = tb[row + CH];
            Bfr[k][0] = p0 - q0;
            Bfr[k][1] = p1 - q1;
        }

        const bool count = (t0 >= t_acc);   // wave-uniform

        #pragma unroll
        for (int al = 0; al < 3; ++al) {
            // seed accumulator with the rank-1 carry term
            v8f D;
            #pragma unroll
            for (int i = 0; i < 8; ++i) D[i] = dec[al][i] * carry[al];
            // E = sum_k A_k @ B_k + C  (4 chained f32 WMMAs)
            #pragma unroll
            for (int k = 0; k < 4; ++k) {
                D = __builtin_amdgcn_wmma_f32_16x16x4_f32(
                        false, Afr[al][k], false, Bfr[k],
                        (short)0, D, false, false);
            }
            // carry = E[row 15, col n] : VGPR7 of lane 16+n
            carry[al] = __shfl(D[7], 16 + n, 32);
            if (count) {
                float s = 0.0f;
                #pragma unroll
                for (int i = 0; i < 8; ++i) s += D[i] * D[i];
                acc += weights[al] * s;
            }
        }
    }

    // wave reduce
    #pragma unroll
    for (int off = 16; off > 0; off >>= 1)
        acc += __shfl_xor(acc, off, 32);

    __shared__ float sred[WAVES_PER_BLOCK];
    if (lane == 0) sred[wib] = acc;
    __syncthreads();
    if (threadIdx.x == 0) {
        float s = 0.0f;
        #pragma unroll
        for (int i = 0; i < WAVES_PER_BLOCK; ++i) s += sred[i];
        partials[blockIdx.x] = s;
    }
}

__global__ __launch_bounds__(256)
void MultiScaleTrendLoss_final_reduce(const float* __restrict__ partials,
                                      float* __restrict__ out)
{
    __shared__ float s[256];
    float v = 0.0f;
    for (int i = threadIdx.x; i < NBLOCKS; i += 256) v += partials[i];
    s[threadIdx.x] = v;
    __syncthreads();
    for (int off = 128; off > 0; off >>= 1) {
        if ((int)threadIdx.x < off) s[threadIdx.x] += s[threadIdx.x + off];
        __syncthreads();
    }
    if (threadIdx.x == 0)
        out[0] = s[0] * (1.0f / 16777216.0f);   // / (64*4096*64)
}

extern "C" void kernel_launch(void* const* d_in, const int* in_sizes, int n_in,
                              void* d_out, int out_size, void* d_ws, size_t ws_size,
                              hipStream_t stream) {
    const float* pred   = (const float*)d_in[0];
    const float* target = (const float*)d_in[1];
    float* partials = (float*)d_ws;          // NBLOCKS floats, fully overwritten
    float* out      = (float*)d_out;

    MultiScaleTrendLoss_ema_wmma_kernel<<<NBLOCKS, BLOCK_THREADS, 0, stream>>>(
        pred, target, partials);
    MultiScaleTrendLoss_final_reduce<<<1, 256, 0, stream>>>(partials, out);
}